// BPMLLLoss_27281632264919
// MI455X (gfx1250) — compile-verified
//
#include <hip/hip_runtime.h>

// BP-MLL loss, factorized form:
//   pair_sum[b] = (sum_{t=1} exp(-x)) * (sum_{t=0} exp(+x))
//   y_norm[b]   = n_pos * (C - n_pos)
//   out         = (1/B) * sum_b pair_sum[b] / y_norm[b]
// B = 32, C = 2048 (fixed by the reference setup).

typedef __attribute__((ext_vector_type(2))) float v2f;
typedef __attribute__((ext_vector_type(8))) float v8f;

#define BPMLL_B 32
#define BPMLL_C 2048

// Cross-lane (32-lane) sum using the f32 WMMA pipe.
// A[m,k] = p[lane m + 16*(k>=2)] (both A components carry p, so the result is
// independent of the exact K<->VGPR striping); B = all ones.
// D[m,n] = 2*(p[m] + p[m+16]).  Per-lane sum of the 8 D VGPRs gives the
// half-sums; one shfl_xor(16) completes:  s = 2 * sum_{l=0..31} p[l].
__device__ __forceinline__ float wave_reduce_wmma(float p) {
    v2f a; a.x = p;    a.y = p;
    v2f b; b.x = 1.0f; b.y = 1.0f;
    v8f c = {};
    v8f d = __builtin_amdgcn_wmma_f32_16x16x4_f32(
        /*neg_a=*/false, a, /*neg_b=*/false, b,
        /*c_mod=*/(short)0, c, /*reuse_a=*/false, /*reuse_b=*/false);
    float s = d[0] + d[1] + d[2] + d[3] + d[4] + d[5] + d[6] + d[7];
    s += __shfl_xor(s, 16, 32);
    return 0.5f * s;
}

// One wave per row: thread-local accumulation over 64 elements/lane with
// float4/int4 coalesced loads, then 3 WMMA cross-lane reductions.
__global__ void __launch_bounds__(256)
bpmll_row_kernel(const float* __restrict__ input,
                 const int* __restrict__ target,
                 float* __restrict__ row_loss) {
    const int lane = threadIdx.x & 31;
    const int wave = threadIdx.x >> 5;
    const int row  = blockIdx.x * (blockDim.x >> 5) + wave;   // 0..31

    const float* __restrict__ xr = input  + row * BPMLL_C;
    const int*   __restrict__ tr = target + row * BPMLL_C;

    float pe = 0.0f;   // sum of exp(-x) over positives
    float ne = 0.0f;   // sum of exp(+x) over negatives
    float pc = 0.0f;   // count of positives (exact in f32)

#pragma unroll
    for (int it = 0; it < BPMLL_C / (32 * 4); ++it) {
        const int idx = (it * 32 + lane) * 4;                 // 512B/wave/iter
        const float4 x = *reinterpret_cast<const float4*>(xr + idx);
        const int4   t = *reinterpret_cast<const int4*>(tr + idx);

        {   const bool p = (t.x == 1);
            const float e = __expf(p ? -x.x : x.x);
            pe += p ? e : 0.0f;  ne += p ? 0.0f : e;  pc += p ? 1.0f : 0.0f; }
        {   const bool p = (t.y == 1);
            const float e = __expf(p ? -x.y : x.y);
            pe += p ? e : 0.0f;  ne += p ? 0.0f : e;  pc += p ? 1.0f : 0.0f; }
        {   const bool p = (t.z == 1);
            const float e = __expf(p ? -x.z : x.z);
            pe += p ? e : 0.0f;  ne += p ? 0.0f : e;  pc += p ? 1.0f : 0.0f; }
        {   const bool p = (t.w == 1);
            const float e = __expf(p ? -x.w : x.w);
            pe += p ? e : 0.0f;  ne += p ? 0.0f : e;  pc += p ? 1.0f : 0.0f; }
    }

    const float PE = wave_reduce_wmma(pe);
    const float NE = wave_reduce_wmma(ne);
    const float PC = wave_reduce_wmma(pc);

    if (lane == 0) {
        const float NC = (float)BPMLL_C - PC;
        row_loss[row] = (PE * NE) / (PC * NC);
    }
}

// Single wave: reduce the 32 per-row losses (WMMA again) and scale by 1/B.
__global__ void __launch_bounds__(32)
bpmll_final_kernel(const float* __restrict__ row_loss,
                   float* __restrict__ out) {
    const int lane = threadIdx.x & 31;
    const float v = row_loss[lane];
    const float s = wave_reduce_wmma(v);
    if (lane == 0) {
        out[0] = s * (1.0f / (float)BPMLL_B);
    }
}

extern "C" void kernel_launch(void* const* d_in, const int* in_sizes, int n_in,
                              void* d_out, int out_size, void* d_ws, size_t ws_size,
                              hipStream_t stream) {
    const float* input  = (const float*)d_in[0];   // [32, 2048] f32
    const int*   target = (const int*)d_in[1];     // [32, 2048] i32
    float* ws  = (float*)d_ws;                     // 32 per-row losses
    float* out = (float*)d_out;                    // scalar f32

    // 4 blocks x 8 waves = 32 waves = one wave per row; EXEC all-ones at WMMA.
    bpmll_row_kernel<<<dim3(4), dim3(256), 0, stream>>>(input, target, ws);
    bpmll_final_kernel<<<dim3(1), dim3(32), 0, stream>>>(ws, out);
}